// SpatialGate_14250701488756
// MI455X (gfx1250) — compile-verified
//
#include <hip/hip_runtime.h>
#include <hip/hip_bf16.h>
#include <stdint.h>

typedef float v2f __attribute__((ext_vector_type(2)));
typedef float v4f __attribute__((ext_vector_type(4)));
typedef float v8f __attribute__((ext_vector_type(8)));

#define NB   32      // batch
#define NC   256     // channels
#define NH   56
#define NW   56
#define NHW  3136    // 56*56
#define NHWG 784     // NHW/4

// ---------------------------------------------------------------------------
// Pass 1: channel-wise max & mean.  x[b,c,h,w] -> planes[b][0]=max, [b][1]=avg
// HBM-bound: reads 103 MB, writes 0.8 MB.  float4 per thread, coalesced.
// ---------------------------------------------------------------------------
__global__ void __launch_bounds__(256) k_reduce(const float* __restrict__ x,
                                                float* __restrict__ planes) {
  int t = blockIdx.x * 256 + threadIdx.x;          // 0 .. 25087 (exact)
  int b = t / NHWG;
  int g = t - b * NHWG;
  const v4f* px = (const v4f*)x + (size_t)b * NC * NHWG + g;
  v4f v = px[0];
  v4f mx = v;
  v4f sm = v;
  #pragma unroll 4
  for (int c = 1; c < NC; ++c) {
    v4f u = px[(size_t)c * NHWG];
    mx.x = fmaxf(mx.x, u.x); mx.y = fmaxf(mx.y, u.y);
    mx.z = fmaxf(mx.z, u.z); mx.w = fmaxf(mx.w, u.w);
    sm += u;
  }
  v4f av = sm * (1.0f / (float)NC);
  v4f* pp = (v4f*)planes;
  pp[(size_t)(b * 2 + 0) * NHWG + g] = mx;
  pp[(size_t)(b * 2 + 1) * NHWG + g] = av;
}

// ---------------------------------------------------------------------------
// Pass 2: 7x7 conv (2ch -> 1ch, pad 3) + BN(eval) + sigmoid, via WMMA f32
// 16x16x4.  One wave per block computes a 16x16 output tile as a GEMM:
//   D[m,n] = sum_{k=(c,dy,j)} tile[c][m+dy][j] * w[c][dy][j-n]
// K padded to 14 pairs * 24 j-slots = 336 = 84 WMMA steps of K=4; the pad
// slots (j=22,23) read zeroed LDS columns and a zero Toeplitz weight, so
// they contribute nothing.  K ordering k = (c*7+dy)*24 + j keeps each WMMA
// step inside one (c,dy) pair -> pure incremental indexing, no divisions.
// Input strip (2 x 22 x 22) staged into LDS via global_load_async_to_lds_b32
// (borders pre-zeroed with DS stores).
// ---------------------------------------------------------------------------
__global__ void __launch_bounds__(32) k_conv_gate(const float* __restrict__ planes,
                                                  const float* __restrict__ wgt,
                                                  const float* __restrict__ gma,
                                                  const float* __restrict__ bta,
                                                  const float* __restrict__ rmean,
                                                  const float* __restrict__ rvar,
                                                  float* __restrict__ gate) {
  __shared__ float tile[2][22][24];   // row stride 24 (pad cols 22,23 = 0)
  __shared__ float wsm[112];          // 98 weights used

  const int lane = threadIdx.x;       // 0..31, single wave32
  const int x0 = blockIdx.x * 16;
  const int y0 = blockIdx.y * 16;
  const int b  = blockIdx.z;

  // Zero-fill tile (covers halo padding AND the j=22,23 K-pad columns).
  float* tf = &tile[0][0][0];
  for (int i = lane; i < 2 * 22 * 24; i += 32) tf[i] = 0.0f;
  for (int i = lane; i < 112; i += 32) wsm[i] = (i < 98) ? wgt[i] : 0.0f;
  // DS stores must land before the async engine overwrites the interior.
  asm volatile("s_wait_dscnt 0x0" ::: "memory");

  // Async copy the in-bounds interior: 2 * 22 * 22 = 968 elements.
  for (int i = lane; i < 968; i += 32) {
    int c  = i / 484;
    int r  = (i - c * 484) / 22;
    int cc = i - c * 484 - r * 22;
    int gy = y0 + r - 3;
    int gx = x0 + cc - 3;
    if (gy >= 0 && gy < NH && gx >= 0 && gx < NW) {
      uint32_t la = (uint32_t)(uintptr_t)&tile[c][r][cc];   // LDS byte offset
      uint64_t ga = (uint64_t)(uintptr_t)(planes +
                    ((size_t)(b * 2 + c)) * NHW + gy * NW + gx);
      asm volatile("global_load_async_to_lds_b32 %0, %1, off"
                   :: "v"(la), "v"(ga) : "memory");
    }
  }
  asm volatile("s_wait_asynccnt 0x0" ::: "memory");
  __syncthreads();

  const int half = lane >> 4;   // A/B K-half: lanes 0-15 hold K{0,1}, 16-31 K{2,3}
  const int m = lane & 15;      // A row
  const int n = m;              // B/D column
  const int jh = half << 1;     // per-lane j offset within a K=4 step

  v8f acc = {0.f, 0.f, 0.f, 0.f, 0.f, 0.f, 0.f, 0.f};
  for (int c = 0; c < 2; ++c) {
    for (int dy = 0; dy < 7; ++dy) {
      const float* trow = &tile[c][m + dy][0];      // lane's A row in LDS
      const float* wrow = &wsm[(c * 7 + dy) * 7];   // 7 taps for this (c,dy)
      #pragma unroll
      for (int s = 0; s < 6; ++s) {                 // 6 K=4 steps over j=0..23
        v2f a{}, bm{};
        #pragma unroll
        for (int q = 0; q < 2; ++q) {
          int j = 4 * s + jh + q;
          a[q] = trow[j];                           // adjacent -> ds_load_b64
          int d  = j - n;                           // dx tap index
          int dc = d < 0 ? 0 : (d > 6 ? 6 : d);
          float wv = wrow[dc];
          bm[q] = (d == dc) ? wv : 0.0f;            // Toeplitz weight
        }
        // 8 args: (neg_a, A, neg_b, B, c_mod, C, reuse_a, reuse_b)
        acc = __builtin_amdgcn_wmma_f32_16x16x4_f32(
            false, a, false, bm, (short)0, acc, false, false);
      }
    }
  }

  const float gamma = gma[0], beta = bta[0], mean = rmean[0], var = rvar[0];
  const float inv = gamma * rsqrtf(var + 1e-5f);
  const int gx = x0 + n;
  // D layout: VGPR r: lanes 0-15 -> M=r, lanes 16-31 -> M=8+r; N = lane&15.
  #pragma unroll
  for (int r = 0; r < 8; ++r) {
    int gy = y0 + (half ? 8 + r : r);
    if (gy < NH && gx < NW) {
      float t = (acc[r] - mean) * inv + beta;
      gate[(size_t)b * NHW + gy * NW + gx] = 1.0f / (1.0f + expf(-t));
    }
  }
}

// ---------------------------------------------------------------------------
// Pass 3: out = x * gate (gate broadcast over 256 channels).
// x re-read should hit L2 (103 MB < 192 MB); output stores are non-temporal
// so the 103 MB written does not evict x/gate from L2.
// ---------------------------------------------------------------------------
__global__ void __launch_bounds__(256) k_gate_mul(const float* __restrict__ x,
                                                  const float* __restrict__ gate,
                                                  float* __restrict__ out) {
  size_t t = (size_t)blockIdx.x * 256 + threadIdx.x;   // float4-group index
  int b = (int)(t / (NC * (size_t)NHWG));
  size_t rem = t - (size_t)b * (NC * (size_t)NHWG);
  int g = (int)(rem % NHWG);
  v4f xv = ((const v4f*)x)[t];
  v4f gv = ((const v4f*)gate)[(size_t)b * NHWG + g];
  v4f o = xv * gv;
  __builtin_nontemporal_store(o, (v4f*)out + t);
}

// ---------------------------------------------------------------------------
extern "C" void kernel_launch(void* const* d_in, const int* in_sizes, int n_in,
                              void* d_out, int out_size, void* d_ws, size_t ws_size,
                              hipStream_t stream) {
  (void)in_sizes; (void)n_in; (void)out_size; (void)ws_size;
  const float* x     = (const float*)d_in[0];
  const float* w     = (const float*)d_in[1];
  const float* gamma = (const float*)d_in[2];
  const float* beta  = (const float*)d_in[3];
  const float* rmean = (const float*)d_in[4];
  const float* rvar  = (const float*)d_in[5];
  float* out    = (float*)d_out;
  float* planes = (float*)d_ws;                          // 32*2*3136 floats
  float* gate   = planes + (size_t)NB * 2 * NHW;         // 32*3136 floats

  // Pass 1: 32*784 = 25088 threads exactly.
  k_reduce<<<98, 256, 0, stream>>>(x, planes);

  // Pass 2: 4x4 tiles of 16x16 per batch, one wave per block.
  dim3 g2(4, 4, NB);
  k_conv_gate<<<g2, 32, 0, stream>>>(planes, w, gamma, beta, rmean, rvar, gate);

  // Pass 3: 6,422,528 float4 groups = 25088 blocks * 256 threads exactly.
  k_gate_mul<<<25088, 256, 0, stream>>>(x, gate, out);
}